// NetVLADPool_60653528154408
// MI455X (gfx1250) — compile-verified
//
#include <hip/hip_runtime.h>
#include <hip/hip_bf16.h>
#include <math.h>

// NetVLAD fused pool for MI455X (gfx1250, wave32).
//   Pass 1 (netvlad_accum): double-buffered async staging of 16-token x-tiles
//     directly into LDS (global_load_async_to_lds_b128, ASYNCcnt), then
//     GEMM1 logits = x*W^T  via v_wmma_f32_16x16x4_f32,
//     softmax(+bias, mask) in LDS,
//     GEMM2 vlad += assign^T*x via v_wmma_f32_16x16x4_f32 (16 v8f accumulators),
//     flush with global f32 atomics. x (128 MB) is read exactly once -> ~5.5us
//     HBM floor at 23.3 TB/s.
//   Pass 2 (netvlad_norm): subtract asum*centroids, intra-cluster L2 norm,
//     global L2 norm, write output.

typedef __attribute__((ext_vector_type(2))) float v2f;
typedef __attribute__((ext_vector_type(4))) float v4f;
typedef __attribute__((ext_vector_type(8))) float v8f;
typedef __attribute__((ext_vector_type(4))) int   v4i;

#define B_   32
#define N_   4096
#define D_   256
#define K_   16
#define TPW  128   // tokens per workgroup
#define TT   16    // token tile (WMMA M)
#define NTILES (TPW / TT)
#define XS   260   // padded LDS row stride (floats) for x and W (1040B = 65*16)
#define AS   20    // padded LDS row stride for assign^T

#define GLOBAL_AS __attribute__((address_space(1)))
#define LDS_AS    __attribute__((address_space(3)))

#if defined(__has_builtin)
#if __has_builtin(__builtin_amdgcn_global_load_async_to_lds_b128)
#define HAVE_ASYNC_B128 1
#endif
#if __has_builtin(__builtin_amdgcn_s_wait_asynccnt)
#define HAVE_WAIT_ASYNC 1
#endif
#endif

// Issue one 16-byte global -> LDS async copy for this lane.
// Builtin signature (from clang diagnostic): (v4i AS1*, v4i AS3*, Imm, Imm).
static __device__ __forceinline__ void async_copy_b128(const float* g, float* l) {
#if defined(HAVE_ASYNC_B128)
  __builtin_amdgcn_global_load_async_to_lds_b128(
      (GLOBAL_AS v4i*)(void*)g, (LDS_AS v4i*)(void*)l, 0, 0);
#else
  // ISA: VDST = per-lane LDS byte offset, VADDR = 64-bit global address.
  unsigned loff = (unsigned)(unsigned long long)(void*)l;  // low 32b = LDS offset
  unsigned long long ga = (unsigned long long)g;
  asm volatile("global_load_async_to_lds_b128 %0, %1, off"
               :: "v"(loff), "v"(ga) : "memory");
#endif
}

static __device__ __forceinline__ void async_wait0() {
#if defined(HAVE_WAIT_ASYNC)
  __builtin_amdgcn_s_wait_asynccnt(0);
#else
  asm volatile("s_wait_asynccnt 0" ::: "memory");
#endif
}

__global__ __launch_bounds__(32) void netvlad_accum(
    const float* __restrict__ x, const unsigned char* __restrict__ mask,
    const float* __restrict__ W, const float* __restrict__ bias,
    float* __restrict__ vlad_acc, float* __restrict__ asum_acc)
{
  __shared__ float lds_w[K_ * XS];          // W[k][d], padded rows
  __shared__ float lds_x[2][TT * XS];       // double-buffered x tile [tok][d]
  __shared__ float lds_logits[TT * 17];     // [tok][k]
  __shared__ float lds_at[K_ * AS];         // assign^T [k][tok]
  __shared__ float lds_b[K_];

  const int lane = threadIdx.x;   // 0..31 (wave32)
  const int hf   = lane >> 4;     // half-wave selector
  const int l16  = lane & 15;
  const int batch = blockIdx.y;
  const int n0    = blockIdx.x * TPW;

  // Async-stage W (16x256 = 16 KB) into LDS once (L2-resident source).
  #pragma unroll
  for (int i = 0; i < 32; ++i) {                 // 1024 float4 / 32 lanes
    int idx = i * 32 + lane;
    int row = idx >> 6;                          // 64 float4 per row
    int c4  = idx & 63;
    async_copy_b128(W + row * D_ + c4 * 4, &lds_w[row * XS + c4 * 4]);
  }
  if (lane < K_) lds_b[lane] = bias[lane];

  // Async-stage tile 0 into buffer 0 (prologue of the double buffer).
  {
    const float* xg = x + ((size_t)batch * N_ + n0) * D_;
    #pragma unroll
    for (int i = 0; i < 32; ++i) {
      int idx = i * 32 + lane;
      int row = idx >> 6;
      int c4  = idx & 63;
      async_copy_b128(xg + row * D_ + c4 * 4, &lds_x[0][row * XS + c4 * 4]);
    }
  }

  v8f acc[16];                                   // vlad 16k x 256d in VGPRs
  #pragma unroll
  for (int j = 0; j < 16; ++j) acc[j] = {};
  float asum_r = 0.f;

  int p = 0;
  for (int tile = 0; tile < NTILES; ++tile) {
    async_wait0();               // buffer p (and W on first iter) now resident
    __syncthreads();

    // Prefetch next tile into the other buffer while we compute on this one.
    if (tile + 1 < NTILES) {
      const float* xg = x + ((size_t)batch * N_ + n0 + (tile + 1) * TT) * D_;
      float* dst = &lds_x[p ^ 1][0];
      #pragma unroll
      for (int i = 0; i < 32; ++i) {
        int idx = i * 32 + lane;
        int row = idx >> 6;
        int c4  = idx & 63;
        async_copy_b128(xg + row * D_ + c4 * 4, dst + row * XS + c4 * 4);
      }
    }

    const float* xb = &lds_x[p][0];
    const int tok0 = n0 + tile * TT;

    // ---- GEMM1: logits(16 tok x 16 k) = x_tile * W^T, K-loop over d in 4s.
    v8f clog = {};
    #pragma unroll 8
    for (int c = 0; c < D_ / 4; ++c) {
      // A frag: x[tok=l16][4c + 2*hf + {0,1}]
      v2f a  = *(const v2f*)(&xb[l16 * XS + 4 * c + 2 * hf]);
      // B frag: W[k=l16][4c + 2*hf + {0,1}]  (B row = d-within-chunk)
      v2f bw = *(const v2f*)(&lds_w[l16 * XS + 4 * c + 2 * hf]);
      clog = __builtin_amdgcn_wmma_f32_16x16x4_f32(
          false, a, false, bw, (short)0, clog, false, false);
    }
    // C layout: lane l, reg r -> token = r + 8*hf, cluster = l16.
    #pragma unroll
    for (int r = 0; r < 8; ++r)
      lds_logits[(r + 8 * hf) * 17 + l16] = clog[r];
    __syncthreads();

    // ---- softmax over K with bias + mask (lane t handles token t).
    if (lane < TT) {
      int t = lane;
      unsigned char mk = mask[(size_t)batch * N_ + tok0 + t];  // jax bool = 1B
      float li[K_];
      float mx = -3.0e38f;
      #pragma unroll
      for (int k = 0; k < K_; ++k) {
        li[k] = lds_logits[t * 17 + k] + lds_b[k];
        mx = fmaxf(mx, li[k]);
      }
      float s = 0.f;
      #pragma unroll
      for (int k = 0; k < K_; ++k) { li[k] = __expf(li[k] - mx); s += li[k]; }
      // masked token: softmax(-1e9 row) * 0 == 0 in the reference
      float inv = mk ? (1.f / s) : 0.f;
      #pragma unroll
      for (int k = 0; k < K_; ++k) lds_at[k * AS + t] = li[k] * inv;
    }
    __syncthreads();

    // ---- asum[k] += sum_t assign[t][k]  (lane k owns cluster k)
    if (lane < K_) {
      float s = 0.f;
      #pragma unroll
      for (int t = 0; t < TT; ++t) s += lds_at[lane * AS + t];
      asum_r += s;
    }

    // ---- GEMM2: vlad(16k x 256d) += assign^T * x_tile, K over tokens in 4s.
    #pragma unroll
    for (int c2 = 0; c2 < 4; ++c2) {
      // A frag: assignT[k=l16][4c2 + 2*hf + {0,1}]
      v2f af = *(const v2f*)(&lds_at[l16 * AS + 4 * c2 + 2 * hf]);
      #pragma unroll
      for (int j = 0; j < 16; ++j) {             // 16 d-tiles of 16
        v2f bf;
        bf.x = xb[(4 * c2 + 2 * hf)     * XS + 16 * j + l16];
        bf.y = xb[(4 * c2 + 2 * hf + 1) * XS + 16 * j + l16];
        acc[j] = __builtin_amdgcn_wmma_f32_16x16x4_f32(
            false, af, false, bf, (short)0, acc[j], false, false);
      }
    }
    __syncthreads();  // lds_at / lds_x[p] reads done before reuse
    p ^= 1;
  }

  // Flush: acc[j][r] = vlad[k = r + 8*hf][d = 16*j + l16]
  float* vb = vlad_acc + (size_t)batch * (K_ * D_);
  #pragma unroll
  for (int j = 0; j < 16; ++j) {
    #pragma unroll
    for (int r = 0; r < 8; ++r)
      atomicAdd(&vb[(r + 8 * hf) * D_ + 16 * j + l16], acc[j][r]);
  }
  if (lane < K_) atomicAdd(&asum_acc[batch * K_ + lane], asum_r);
}

__global__ __launch_bounds__(256) void netvlad_norm(
    const float* __restrict__ vlad_acc, const float* __restrict__ asum_acc,
    const float* __restrict__ centroids, float* __restrict__ out)
{
  __shared__ float red[8];
  __shared__ float gscale;
  const int b = blockIdx.x;
  const int t = threadIdx.x;      // 256 threads: t>>4 = cluster, t&15 = d-group
  const int k = t >> 4;
  const int l = t & 15;

  const float a = asum_acc[b * K_ + k];
  float v[16];
  float ss = 0.f;
  #pragma unroll
  for (int i = 0; i < 16; ++i) {
    int d = l * 16 + i;
    float val = vlad_acc[(size_t)b * (K_ * D_) + k * D_ + d]
              - a * centroids[k * D_ + d];
    v[i] = val;
    ss += val * val;
  }
  // intra-cluster norm: reduce over the 16 threads sharing this k
  #pragma unroll
  for (int m = 1; m < 16; m <<= 1) ss += __shfl_xor(ss, m, 16);
  float inv = 1.f / fmaxf(sqrtf(ss), 1e-12f);

  float gs = 0.f;
  #pragma unroll
  for (int i = 0; i < 16; ++i) { v[i] *= inv; gs += v[i] * v[i]; }

  // global norm: block-wide reduction (8 waves of 32)
  #pragma unroll
  for (int m = 1; m < 32; m <<= 1) gs += __shfl_xor(gs, m, 32);
  if ((t & 31) == 0) red[t >> 5] = gs;
  __syncthreads();
  if (t == 0) {
    float s = 0.f;
    for (int w = 0; w < 8; ++w) s += red[w];
    gscale = 1.f / fmaxf(sqrtf(s), 1e-12f);
  }
  __syncthreads();
  float gi = gscale;
  #pragma unroll
  for (int i = 0; i < 16; ++i)
    out[(size_t)b * (K_ * D_) + k * D_ + l * 16 + i] = v[i] * gi;
}

__global__ void zero_ws_kernel(float* p, int n) {
  int i = blockIdx.x * blockDim.x + threadIdx.x;
  if (i < n) p[i] = 0.f;
}

extern "C" void kernel_launch(void* const* d_in, const int* in_sizes, int n_in,
                              void* d_out, int out_size, void* d_ws, size_t ws_size,
                              hipStream_t stream) {
  const float*         x         = (const float*)d_in[0];
  const unsigned char* mask      = (const unsigned char*)d_in[1];  // jax bool
  const float*         W         = (const float*)d_in[2];
  const float*         bias      = (const float*)d_in[3];
  const float*         centroids = (const float*)d_in[4];
  float*               out       = (float*)d_out;

  float* vlad_acc = (float*)d_ws;                 // 32 * 4096 floats
  float* asum_acc = vlad_acc + B_ * K_ * D_;      // 32 * 16 floats
  const int ztotal = B_ * K_ * D_ + B_ * K_;

  zero_ws_kernel<<<(ztotal + 255) / 256, 256, 0, stream>>>(vlad_acc, ztotal);

  dim3 grid(N_ / TPW, B_);                        // 32 x 32 = 1024 workgroups
  netvlad_accum<<<grid, 32, 0, stream>>>(x, mask, W, bias, vlad_acc, asum_acc);

  netvlad_norm<<<B_, 256, 0, stream>>>(vlad_acc, asum_acc, centroids, out);
}